// BoneLengthLoss_24043226923676
// MI455X (gfx1250) — compile-verified
//
#include <hip/hip_runtime.h>

#ifndef __has_builtin
#define __has_builtin(x) 0
#endif

#if __has_builtin(__builtin_amdgcn_global_load_async_to_lds_b128)
#define HAVE_ASYNC_B128 1
#else
#define HAVE_ASYNC_B128 0
#endif

// Builtin parameter types (probe-confirmed in round 1/2): generic pointer to
// ext-vector int4 for the global source; AS(3) typed pointer for the LDS dest.
typedef int v4i __attribute__((ext_vector_type(4)));
typedef __attribute__((address_space(3))) v4i* lds_v4i;

// Generic LDS pointer's low 32 bits are the LDS byte offset, so the
// integer round-trip cast to a 32-bit AS(3) pointer is numerically correct.
#define ASYNC_LOAD_B128(gsrc, ldst)                        \
    __builtin_amdgcn_global_load_async_to_lds_b128(        \
        (v4i*)(gsrc), (lds_v4i)(unsigned long long)(ldst), 0, 0)

#if __has_builtin(__builtin_amdgcn_s_wait_asynccnt)
#define WAIT_ASYNC_0() __builtin_amdgcn_s_wait_asynccnt(0)
#else
#define WAIT_ASYNC_0() asm volatile("s_wait_asynccnt 0" ::: "memory")
#endif

// 64 poses per 256-thread block; 63 floats per pose per tensor.
constexpr int BPB    = 64;
constexpr int JF     = 63;          // floats per pose (21 joints * 3)
constexpr int FLOATS = BPB * JF;    // 4032 floats staged per tensor
constexpr int V4     = FLOATS / 4;  // 1008 float4 per tensor per block

__global__ __launch_bounds__(256) void bone_length_loss_kernel(
    const float* __restrict__ jout,
    const float* __restrict__ jgt,
    float* __restrict__ out,
    int nBatch)
{
    __shared__ float4 s_o4[V4];
    __shared__ float4 s_g4[V4];
    __shared__ float  s_res[BPB];

    const int       tid    = threadIdx.x;
    const long long f4base = (long long)blockIdx.x * V4;
    const long long totF4  = ((long long)nBatch * JF) / 4;
    const long long b0     = (long long)blockIdx.x * BPB;

    const float4* go = (const float4*)jout + f4base;
    const float4* gg = (const float4*)jgt  + f4base;

    // ---- Stage 2 x 16 KB of pose data through LDS, coalesced async B128 ----
    if (f4base + V4 <= totF4) {
        // Full block (the only case for B = 1M): unrolled burst of 8 async
        // copies in flight before a single ASYNCcnt wait. 1008 = 3*256 + 240,
        // so only the last trip carries an exec predicate.
#pragma unroll
        for (int k = 0; k < 4; ++k) {
            const int i = k * 256 + tid;
            if (i < V4) {
#if HAVE_ASYNC_B128
                ASYNC_LOAD_B128(go + i, &s_o4[i]);
                ASYNC_LOAD_B128(gg + i, &s_g4[i]);
#else
                s_o4[i] = go[i];
                s_g4[i] = gg[i];
#endif
            }
        }
    } else {
        // Rare partial tail block: per-element guard.
        for (int i = tid; i < V4; i += 256) {
            if (f4base + i < totF4) {
#if HAVE_ASYNC_B128
                ASYNC_LOAD_B128(go + i, &s_o4[i]);
                ASYNC_LOAD_B128(gg + i, &s_g4[i]);
#else
                s_o4[i] = go[i];
                s_g4[i] = gg[i];
#endif
            }
        }
    }
#if HAVE_ASYNC_B128
    WAIT_ASYNC_0();
#endif
    __syncthreads();

    // ---- Compute: 4 lanes per pose, 5 bones each ----
    const float* so = (const float*)s_o4;
    const float* sg = (const float*)s_g4;

    const int bl  = tid >> 2;   // local pose 0..63
    const int sub = tid & 3;    // 0..3

    const float* jo = so + bl * JF;
    const float* jg = sg + bl * JF;

    float acc = 0.0f;
#pragma unroll
    for (int k = 0; k < 5; ++k) {
        const int m = sub * 5 + k;            // bone index 0..19
        const int a = ((m & 3) == 0) ? 0 : m; // skeleton[:,0]
        const int b = m + 1;                  // skeleton[:,1]

        const float dox = jo[a * 3 + 0] - jo[b * 3 + 0];
        const float doy = jo[a * 3 + 1] - jo[b * 3 + 1];
        const float doz = jo[a * 3 + 2] - jo[b * 3 + 2];
        const float lo  = __builtin_sqrtf(__builtin_fmaf(dox, dox,
                              __builtin_fmaf(doy, doy, doz * doz)));

        const float dgx = jg[a * 3 + 0] - jg[b * 3 + 0];
        const float dgy = jg[a * 3 + 1] - jg[b * 3 + 1];
        const float dgz = jg[a * 3 + 2] - jg[b * 3 + 2];
        const float lg  = __builtin_sqrtf(__builtin_fmaf(dgx, dgx,
                              __builtin_fmaf(dgy, dgy, dgz * dgz)));

        const float d = lo - lg;
        acc = __builtin_fmaf(d, d, acc);
    }

    // Reduce across the 4 collaborating lanes (consecutive lanes of a wave32).
    acc += __shfl_xor(acc, 1);
    acc += __shfl_xor(acc, 2);

    if (sub == 0) s_res[bl] = acc;
    __syncthreads();

    // ---- Coalesced 256 B non-temporal output store per block ----
    if (tid < BPB && (b0 + tid) < (long long)nBatch) {
        __builtin_nontemporal_store(s_res[tid], &out[b0 + tid]);
    }
}

extern "C" void kernel_launch(void* const* d_in, const int* in_sizes, int n_in,
                              void* d_out, int out_size, void* d_ws, size_t ws_size,
                              hipStream_t stream) {
    const float* jout = (const float*)d_in[0];
    const float* jgt  = (const float*)d_in[1];
    float*       out  = (float*)d_out;

    const int nBatch = in_sizes[0] / JF;          // B = 1,048,576
    const int grid   = (nBatch + BPB - 1) / BPB;  // 16,384 blocks

    hipLaunchKernelGGL(bone_length_loss_kernel, dim3(grid), dim3(256), 0, stream,
                       jout, jgt, out, nBatch);
}